// multi_head_GAT_83468394431132
// MI455X (gfx1250) — compile-verified
//
#include <hip/hip_runtime.h>
#include <math.h>

// Problem constants (match reference)
#define HEADS 8
#define DIM   32
#define HD    256     // HEADS*DIM
#define STEPS 3

typedef float v2f __attribute__((ext_vector_type(2)));
typedef float v8f __attribute__((ext_vector_type(8)));

// ---------------------------------------------------------------------------
// float atomic max via sign-partitioned int/uint atomics (native L2 atomics,
// no CAS loop). Correct for all sign combinations; init value must be -inf.
// ---------------------------------------------------------------------------
__device__ __forceinline__ void atomicMaxF(float* addr, float val) {
    if (val >= 0.0f) {
        atomicMax((int*)addr, __float_as_int(val));
    } else {
        atomicMin((unsigned int*)addr, (unsigned int)__float_as_int(val));
    }
}

__device__ __forceinline__ float leaky(float v) {
    return v > 0.0f ? v : 0.01f * v;
}

// ---------------------------------------------------------------------------
// Per-step state init: agg := 0, maxv := -inf, sumv := 0
// ---------------------------------------------------------------------------
__global__ void init_kernel(float* __restrict__ agg, float* __restrict__ maxv,
                            float* __restrict__ sumv, int NHD, int NH) {
    int t = blockIdx.x * blockDim.x + threadIdx.x;
    if (t < NHD) agg[t] = 0.0f;
    if (t < NH) {
        maxv[t] = -INFINITY;
        sumv[t] = 0.0f;
    }
}

// ---------------------------------------------------------------------------
// q = h @ Wq + bq using V_WMMA_F32_16X16X4_F32 (full fp32 precision).
// One wave -> one 16x16 output tile. Bias folded into accumulator init.
// grid = (Nrows/16, 2), block = 256 (8 waves; wave w covers col tile
// blockIdx.y*8+w, i.e. 16 col tiles of 16 = 256 output columns).
//
// fp32 16x16x4 VGPR layouts (ISA 7.12.2):
//   A (16x4):  lanes 0-15 M=lane, v0=K0 v1=K1 ; lanes 16-31 M=lane-16, v0=K2 v1=K3
//   B (4x16):  lanes 0-15 N=lane, v0=K0 v1=K1 ; lanes 16-31 N=lane-16, v0=K2 v1=K3
//   C (16x16): vgpr r: lanes 0-15 -> (M=r, N=lane); lanes 16-31 -> (M=r+8, N=lane-16)
// ---------------------------------------------------------------------------
__global__ __launch_bounds__(256) void gemm_q_wmma(
    const float* __restrict__ h, const float* __restrict__ Wq,
    const float* __restrict__ bq, float* __restrict__ q, int Nrows) {
    const int lane = threadIdx.x & 31;
    const int wave = threadIdx.x >> 5;
    const int rowbase = blockIdx.x * 16;
    const int cb = (blockIdx.y * 8 + wave) * 16;   // column-tile base (0..240)
    if (rowbase >= Nrows) return;

    const int lo = lane & 15;
    const int hi = lane >> 4;   // 0: K pair {0,1}; 1: K pair {2,3}

    v8f c;
    const float bias = bq[cb + lo];
#pragma unroll
    for (int r = 0; r < 8; ++r) c[r] = bias;

    const float* arow = h  + (size_t)(rowbase + lo) * HD + hi * 2;
    const float* bcol = Wq + (size_t)(hi * 2) * HD + cb + lo;

#pragma unroll 4
    for (int k0 = 0; k0 < HD; k0 += 4) {
        v2f a = *(const v2f*)(arow + k0);          // h[row, k0+hi*2 .. +1]
        v2f b;
        b.x = bcol[(size_t)k0 * HD];               // Wq[k0+hi*2,   n]
        b.y = bcol[(size_t)k0 * HD + HD];          // Wq[k0+hi*2+1, n]
        c = __builtin_amdgcn_wmma_f32_16x16x4_f32(
                false, a, false, b, (short)0, c, false, false);
    }

    float* qrow = q + (size_t)(rowbase + hi * 8) * HD + cb + lo;
#pragma unroll
    for (int r = 0; r < 8; ++r) qrow[(size_t)r * HD] = c[r];
}

// ---------------------------------------------------------------------------
// Per-node attention-logit halves:
//   a_snd[n,h] = <q[n,h,:], Wa[0:D]> ;  a_rcv[n,h] = <q[n,h,:], Wa[D:2D]>
// (q[n*HD + h*D + d] == q[(n*H+h)*D + d], so index flat by t = n*H+h)
// ---------------------------------------------------------------------------
__global__ void alpha_kernel(const float* __restrict__ q,
                             const float* __restrict__ Wa,
                             float* __restrict__ a_snd, float* __restrict__ a_rcv,
                             int NH) {
    int t = blockIdx.x * blockDim.x + threadIdx.x;
    if (t >= NH) return;
    const float* qp = q + (size_t)t * DIM;
    float s0 = 0.0f, s1 = 0.0f;
#pragma unroll
    for (int d = 0; d < DIM; ++d) {
        float v = qp[d];
        s0 += v * Wa[d];
        s1 += v * Wa[DIM + d];
    }
    a_snd[t] = s0;
    a_rcv[t] = s1;
}

// ---------------------------------------------------------------------------
// Edge logits + running segment max (per receiver, per head).
// t = e*H + h so 8 consecutive lanes share one edge -> coalesced gathers.
// ---------------------------------------------------------------------------
__global__ void logits_kernel(const int* __restrict__ snd, const int* __restrict__ rcv,
                              const float* __restrict__ a_s, const float* __restrict__ a_r,
                              const float* __restrict__ ba_p,
                              float* __restrict__ wbuf, float* __restrict__ maxv, int EH) {
    int t = blockIdx.x * blockDim.x + threadIdx.x;
    if (t >= EH) return;
    int e = t >> 3, hh = t & 7;
    int sn = snd[e], rn = rcv[e];
    float l = a_s[sn * HEADS + hh] + a_r[rn * HEADS + hh] + ba_p[0];
    wbuf[t] = l;
    atomicMaxF(&maxv[rn * HEADS + hh], l);
}

// ---------------------------------------------------------------------------
// exp(logit - segmax) in place + segment sum.
// ---------------------------------------------------------------------------
__global__ void exp_kernel(const int* __restrict__ rcv, float* __restrict__ wbuf,
                           const float* __restrict__ maxv, float* __restrict__ sumv, int EH) {
    int t = blockIdx.x * blockDim.x + threadIdx.x;
    if (t >= EH) return;
    int e = t >> 3, hh = t & 7;
    int rn = rcv[e];
    float w = __expf(wbuf[t] - maxv[rn * HEADS + hh]);
    wbuf[t] = w;
    atomicAdd(&sumv[rn * HEADS + hh], w);
}

// ---------------------------------------------------------------------------
// agg[rcv,h,:] += (w/sum) * q[snd,h,:]. One thread per (edge, head, 4 dims);
// an 8-thread group covers one (edge,head) -> 128B contiguous load + 128B of
// contiguous atomics, all L2-resident (q and agg each 20 MB << 192 MB L2).
// ---------------------------------------------------------------------------
__global__ void agg_kernel(const int* __restrict__ snd, const int* __restrict__ rcv,
                           const float* __restrict__ wbuf, const float* __restrict__ sumv,
                           const float* __restrict__ q, float* __restrict__ agg, int total) {
    int t = blockIdx.x * blockDim.x + threadIdx.x;
    if (t >= total) return;
    int e  = t >> 6;
    int r  = t & 63;
    int hh = r >> 3;
    int d4 = (r & 7) << 2;
    int sn = snd[e], rn = rcv[e];
    float coef = wbuf[e * HEADS + hh] / sumv[rn * HEADS + hh];
    const float4 qs = *(const float4*)(q + ((size_t)sn * HEADS + hh) * DIM + d4);
    float* ap = agg + ((size_t)rn * HEADS + hh) * DIM + d4;
    atomicAdd(ap + 0, coef * qs.x);
    atomicAdd(ap + 1, coef * qs.y);
    atomicAdd(ap + 2, coef * qs.z);
    atomicAdd(ap + 3, coef * qs.w);
}

// h_next = leaky_relu(agg)   (intermediate steps)
__global__ void relu_kernel(const float* __restrict__ agg, float* __restrict__ hn, int n) {
    int t = blockIdx.x * blockDim.x + threadIdx.x;
    if (t >= n) return;
    hn[t] = leaky(agg[t]);
}

// out = leaky_relu(mean over heads of agg)   (final step, N x DIM)
__global__ void final_kernel(const float* __restrict__ agg, float* __restrict__ out, int ND) {
    int t = blockIdx.x * blockDim.x + threadIdx.x;
    if (t >= ND) return;
    int n = t >> 5, d = t & 31;
    float s = 0.0f;
#pragma unroll
    for (int hh = 0; hh < HEADS; ++hh)
        s += agg[((size_t)n * HEADS + hh) * DIM + d];
    s *= (1.0f / HEADS);
    out[t] = leaky(s);
}

// ---------------------------------------------------------------------------
extern "C" void kernel_launch(void* const* d_in, const int* in_sizes, int n_in,
                              void* d_out, int out_size, void* d_ws, size_t ws_size,
                              hipStream_t stream) {
    const float* nodes     = (const float*)d_in[0];
    const int*   senders   = (const int*)d_in[1];
    const int*   receivers = (const int*)d_in[2];
    const float* Wq        = (const float*)d_in[3];  // (STEPS, HD, HD)
    const float* bq        = (const float*)d_in[4];  // (STEPS, HD)
    const float* Wa        = (const float*)d_in[5];  // (STEPS, 2*DIM, 1)
    const float* ba        = (const float*)d_in[6];  // (STEPS, 1)

    const int N  = in_sizes[0] / HD;   // 20000
    const int E  = in_sizes[1];        // 320000
    const int NH = N * HEADS;
    const int NHD = N * HD;
    const int EH = E * HEADS;

    // Workspace carve-up (floats): q | hbuf | agg | wbuf | a_snd | a_rcv | maxv | sumv
    float* ws    = (float*)d_ws;
    float* q     = ws;                 ws += NHD;   // 5.12M
    float* hbuf  = ws;                 ws += NHD;
    float* agg   = ws;                 ws += NHD;
    float* wbuf  = ws;                 ws += EH;    // 2.56M
    float* a_snd = ws;                 ws += NH;
    float* a_rcv = ws;                 ws += NH;
    float* maxv  = ws;                 ws += NH;
    float* sumv  = ws;                 ws += NH;
    (void)ws_size;

    const int BT = 256;
    dim3 gemm_grid((N + 15) / 16, 2);

    for (int s = 0; s < STEPS; ++s) {
        const float* hcur = (s == 0) ? nodes : hbuf;
        const float* Wqs = Wq + (size_t)s * HD * HD;
        const float* bqs = bq + (size_t)s * HD;
        const float* Was = Wa + (size_t)s * 2 * DIM;
        const float* bas = ba + (size_t)s;

        init_kernel<<<(NHD + BT - 1) / BT, BT, 0, stream>>>(agg, maxv, sumv, NHD, NH);

        gemm_q_wmma<<<gemm_grid, BT, 0, stream>>>(hcur, Wqs, bqs, q, N);

        alpha_kernel<<<(NH + BT - 1) / BT, BT, 0, stream>>>(q, Was, a_snd, a_rcv, NH);

        logits_kernel<<<(EH + BT - 1) / BT, BT, 0, stream>>>(
            senders, receivers, a_snd, a_rcv, bas, wbuf, maxv, EH);

        exp_kernel<<<(EH + BT - 1) / BT, BT, 0, stream>>>(receivers, wbuf, maxv, sumv, EH);

        int total = E * HEADS * (DIM / 4);
        agg_kernel<<<(total + BT - 1) / BT, BT, 0, stream>>>(
            senders, receivers, wbuf, sumv, q, agg, total);

        if (s < STEPS - 1) {
            relu_kernel<<<(NHD + BT - 1) / BT, BT, 0, stream>>>(agg, hbuf, NHD);
        } else {
            int ND = N * DIM;
            final_kernel<<<(ND + BT - 1) / BT, BT, 0, stream>>>(agg, (float*)d_out, ND);
        }
    }
}